// GCN_28836410425874
// MI455X (gfx1250) — compile-verified
//
#include <hip/hip_runtime.h>

typedef float v2f __attribute__((ext_vector_type(2)));
typedef float v4f __attribute__((ext_vector_type(4)));
typedef float v8f __attribute__((ext_vector_type(8)));

#define DH 96        // D_IN == D_HID == 96
#define WPAD 104     // LDS row stride (floats): 2 rows = 208 ≡ +16 banks mod 64 -> conflict-free
#define WAVES_PER_BLOCK 8

// ---------- utility ----------
__global__ void gcn_zero_kernel(float* __restrict__ p, long n) {
  long i = (long)blockIdx.x * blockDim.x + threadIdx.x;
  if (i < n) p[i] = 0.0f;
}

// deg[dst[e]] += 1
__global__ void gcn_deg_kernel(const int* __restrict__ dst, float* __restrict__ deg, int E) {
  int e = blockIdx.x * blockDim.x + threadIdx.x;
  if (e < E) atomicAdd(&deg[dst[e]], 1.0f);
}

// dinv = rsqrt(deg + 1)   (+1 self loop)
__global__ void gcn_dinv_kernel(const float* __restrict__ deg, float* __restrict__ dinv, int N) {
  int i = blockIdx.x * blockDim.x + threadIdx.x;
  if (i < N) dinv[i] = rsqrtf(deg[i] + 1.0f);
}

// ---------- dense GEMM via f32 WMMA: H = X (NxDH) * W (DHxDH) ----------
// 8 waves/block; W staged once per block into LDS; each wave computes a full
// 16x96 output strip with 6 accumulators (A fragment reused across 6 WMMAs).
__global__ void gcn_gemm_wmma_f32(const float* __restrict__ X, const float* __restrict__ W,
                                  float* __restrict__ H, int N) {
  __shared__ float Wlds[DH * WPAD];

  const int tid  = threadIdx.x;
  const int lane = tid & 31;
  const int wid  = tid >> 5;
  const int half = lane >> 4;     // 0: lanes 0-15, 1: lanes 16-31
  const int m    = lane & 15;

  // cooperative LDS fill of W: 96x96 floats as float4, padded row stride
  #pragma unroll
  for (int idx4 = tid; idx4 < (DH * DH) / 4; idx4 += WAVES_PER_BLOCK * 32) {
    const int r  = idx4 / (DH / 4);
    const int c4 = (idx4 % (DH / 4)) * 4;
    *(v4f*)&Wlds[r * WPAD + c4] = *(const v4f*)&W[r * DH + c4];
  }
  __syncthreads();

  const int ntiles = (N + 15) / 16;
  const int tm = blockIdx.x * WAVES_PER_BLOCK + wid;   // 16-row strip index
  if (tm >= ntiles) return;

  int row  = tm * 16 + m;
  int rowc = (row < N) ? row : (N - 1);   // clamp: EXEC must stay all-1s for WMMA
  const float* xr = X + (size_t)rowc * DH;

  v8f c[6];
  #pragma unroll
  for (int t = 0; t < 6; t++) c[t] = (v8f){0.f,0.f,0.f,0.f,0.f,0.f,0.f,0.f};

  #pragma unroll
  for (int kk = 0; kk < DH; kk += 4) {
    const int k0 = kk + 2 * half;
    // A 16x4: lanes0-15 v0=K0,v1=K1 ; lanes16-31 v0=K2,v1=K3
    v2f a;
    a[0] = xr[k0];
    a[1] = xr[k0 + 1];
    #pragma unroll
    for (int t = 0; t < 6; t++) {
      v2f b;
      b[0] = Wlds[k0 * WPAD + t * 16 + m];
      b[1] = Wlds[(k0 + 1) * WPAD + t * 16 + m];
      c[t] = __builtin_amdgcn_wmma_f32_16x16x4_f32(false, a, false, b, (short)0, c[t], false, false);
    }
  }

  // C/D layout: vgpr r -> M = r + 8*half, N = m
  #pragma unroll
  for (int t = 0; t < 6; t++) {
    #pragma unroll
    for (int r = 0; r < 8; r++) {
      int orow = tm * 16 + r + 8 * half;
      if (orow < N) H[(size_t)orow * DH + t * 16 + m] = c[t][r];
    }
  }
}

// ---------- edge scatter: agg[dst] += hlin[src] * dinv[src]*dinv[dst] ----------
// one wave per edge; 32 lanes x 3 features -> coalesced atomic bursts per dst row
__global__ void gcn_scatter_kernel(const float* __restrict__ hlin,
                                   const int* __restrict__ src, const int* __restrict__ dst,
                                   const float* __restrict__ dinv,
                                   float* __restrict__ agg, int E) {
  const int lane = threadIdx.x & 31;
  const int e = blockIdx.x * (blockDim.x >> 5) + (threadIdx.x >> 5);
  if (e >= E) return;
  const int s = src[e];
  const int d = dst[e];
  const float norm = dinv[s] * dinv[d];
  const float* hs = hlin + (size_t)s * DH;
  float* ad = agg + (size_t)d * DH;
  #pragma unroll
  for (int j = 0; j < 3; j++) {
    const int f = lane + 32 * j;
    atomicAdd(&ad[f], hs[f] * norm);
  }
}

// ---------- epilogue: h = relu(agg + hlin*dinv^2 + bias) ----------
__global__ void gcn_finalize_kernel(const float* __restrict__ hlin, const float* __restrict__ agg,
                                    const float* __restrict__ dinv, const float* __restrict__ bias,
                                    float* __restrict__ hout, int N) {
  long i = (long)blockIdx.x * blockDim.x + threadIdx.x;
  if (i >= (long)N * DH) return;
  const int node = (int)(i / DH);
  const int f    = (int)(i % DH);
  const float di = dinv[node];
  const float v  = agg[i] + hlin[i] * di * di + bias[f];
  hout[i] = fmaxf(v, 0.0f);
}

// ---------- global mean pool accumulation ----------
__global__ void gcn_pool_kernel(const float* __restrict__ h, const int* __restrict__ batch,
                                float* __restrict__ pooled, float* __restrict__ cnt, int N) {
  long i = (long)blockIdx.x * blockDim.x + threadIdx.x;
  if (i >= (long)N * DH) return;
  const int node = (int)(i / DH);
  const int f    = (int)(i % DH);
  const int g    = batch[node];
  atomicAdd(&pooled[(size_t)g * DH + f], h[i]);
  if (f == 0) atomicAdd(&cnt[g], 1.0f);
}

// ---------- final FC: out[g][c] = (pooled[g]/cnt[g]) . Wfc[:,c] + bfc[c] ----------
__global__ void gcn_fc_kernel(const float* __restrict__ pooled, const float* __restrict__ cnt,
                              const float* __restrict__ Wfc, const float* __restrict__ bfc,
                              float* __restrict__ out, int C) {
  const int g = blockIdx.x;
  const int c = threadIdx.x;
  if (c >= C) return;
  const float inv = 1.0f / fmaxf(cnt[g], 1.0f);
  float acc = bfc[c];
  const float* pr = pooled + (size_t)g * DH;
  #pragma unroll 8
  for (int f = 0; f < DH; f++) acc = fmaf(pr[f] * inv, Wfc[f * C + c], acc);
  out[g * C + c] = acc;
}

extern "C" void kernel_launch(void* const* d_in, const int* in_sizes, int n_in,
                              void* d_out, int out_size, void* d_ws, size_t ws_size,
                              hipStream_t stream) {
  const float* x     = (const float*)d_in[0];
  const int*   ei    = (const int*)d_in[1];   // [2, E]
  const int*   batch = (const int*)d_in[2];
  const float* W1    = (const float*)d_in[3];
  const float* b1    = (const float*)d_in[4];
  const float* W2    = (const float*)d_in[5];
  const float* b2    = (const float*)d_in[6];
  const float* Wfc   = (const float*)d_in[7];
  const float* bfc   = (const float*)d_in[8];
  float* out = (float*)d_out;

  const int N = in_sizes[2];          // 50000 nodes
  const int E = in_sizes[1] / 2;      // 800000 edges
  const int C = in_sizes[8];          // 10 classes
  const int G = out_size / C;         // 64 graphs
  const int* src = ei;
  const int* dst = ei + E;

  // workspace carve-up
  float* ws = (float*)d_ws;
  size_t o = 0;
  float* deg    = ws + o; o += (size_t)N;
  float* dinv   = ws + o; o += (size_t)N;
  float* hlin   = ws + o; o += (size_t)N * DH;   // linear (pre-aggregation) output
  float* agg    = ws + o; o += (size_t)N * DH;   // scatter accumulator
  float* h      = ws + o; o += (size_t)N * DH;   // post-ReLU activations
  float* pooled = ws + o; o += (size_t)G * DH;
  float* cnt    = ws + o; o += (size_t)G;

  const long NF = (long)N * DH;
  const int ZB = 256;
  const int ntiles = (N + 15) / 16;
  const int gemmBlocks = (ntiles + WAVES_PER_BLOCK - 1) / WAVES_PER_BLOCK;

  // degrees + dinv (shared by both layers)
  gcn_zero_kernel<<<(N + ZB - 1) / ZB, ZB, 0, stream>>>(deg, N);
  gcn_deg_kernel<<<(E + 255) / 256, 256, 0, stream>>>(dst, deg, E);
  gcn_dinv_kernel<<<(N + 255) / 256, 256, 0, stream>>>(deg, dinv, N);

  // ---- layer 1 ----
  gcn_gemm_wmma_f32<<<gemmBlocks, WAVES_PER_BLOCK * 32, 0, stream>>>(x, W1, hlin, N);
  gcn_zero_kernel<<<(int)((NF + ZB - 1) / ZB), ZB, 0, stream>>>(agg, NF);
  gcn_scatter_kernel<<<(E + 7) / 8, 256, 0, stream>>>(hlin, src, dst, dinv, agg, E);
  gcn_finalize_kernel<<<(int)((NF + 255) / 256), 256, 0, stream>>>(hlin, agg, dinv, b1, h, N);

  // ---- layer 2 ----
  gcn_gemm_wmma_f32<<<gemmBlocks, WAVES_PER_BLOCK * 32, 0, stream>>>(h, W2, hlin, N);
  gcn_zero_kernel<<<(int)((NF + ZB - 1) / ZB), ZB, 0, stream>>>(agg, NF);
  gcn_scatter_kernel<<<(E + 7) / 8, 256, 0, stream>>>(hlin, src, dst, dinv, agg, E);
  gcn_finalize_kernel<<<(int)((NF + 255) / 256), 256, 0, stream>>>(hlin, agg, dinv, b2, h, N);

  // ---- mean pool + FC ----
  gcn_zero_kernel<<<((G * DH + G) + ZB - 1) / ZB, ZB, 0, stream>>>(pooled, (long)G * DH + G); // pooled+cnt contiguous
  gcn_pool_kernel<<<(int)((NF + 255) / 256), 256, 0, stream>>>(h, batch, pooled, cnt, N);
  gcn_fc_kernel<<<G, 32, 0, stream>>>(pooled, cnt, Wfc, bfc, out, C);
}